// DevConv_3100966388297
// MI455X (gfx1250) — compile-verified
//
#include <hip/hip_runtime.h>

// ---------------------------------------------------------------------------
// EdgeConv (DGCNN-style) on MI455X, restructured:
//   y = x @ W_theta^T                      (WMMA f32 16x16x4 GEMM)
//   segmax[v] = max_{e: col[e]=v} y[row[e]]   (native f32 max atomics, L2)
//   out = relu((empty ? 0 : segmax - y) @ W_phi^T)   (WMMA GEMM, fused)
// This removes the per-edge GEMM (52 GFLOPs -> 6.6 GFLOPs) and the 819MB
// [E,128] intermediate. y/segmax (51.2MB each) are L2-resident on the 192MB L2.
// ---------------------------------------------------------------------------

typedef __attribute__((ext_vector_type(2))) float v2f;
typedef __attribute__((ext_vector_type(8))) float v8f;

#define NN 100000   // nodes; 100000 = 6250 * 16, no tail
#define CH 128      // in/out channels

// ---------------------------------------------------------------------------
// Kernel 1: Y[m,n] = sum_k X[m,k] * Wt[n,k]   (Y = X @ Wt^T)
// Block: 256 thr = 8 waves. Block -> 16 rows; wave w -> 16-col tile w.
// Per wave: 32 x (2 global_load_b64 + v_wmma_f32_16x16x4_f32).
// A layout (16x4 f32): lane L holds M = L&15; VGPR0/1 = K=(kb,kb+1), kb = (L&16)?2:0
// B layout (4x16 f32): lane L holds N = L&15; same K split (mirrors A).
// D layout: VGPR j holds (M = j + ((L&16)?8:0), N = L&15).
// ---------------------------------------------------------------------------
__global__ __launch_bounds__(256) void gemm_theta_kernel(
    const float* __restrict__ X, const float* __restrict__ Wt,
    float* __restrict__ Y) {
  const int lane = threadIdx.x & 31;
  const int wave = threadIdx.x >> 5;
  const int m0 = blockIdx.x << 4;        // 16-row block
  const int n0 = wave << 4;              // 16-col tile
  const int kb = (lane & 16) ? 2 : 0;

  const float* arow = X + (long long)(m0 + (lane & 15)) * CH + kb;
  const float* brow = Wt + (long long)(n0 + (lane & 15)) * CH + kb;

  v8f acc = {};
#pragma unroll
  for (int kk = 0; kk < CH; kk += 4) {
    v2f a = *(const v2f*)(arow + kk);
    v2f b = *(const v2f*)(brow + kk);
    acc = __builtin_amdgcn_wmma_f32_16x16x4_f32(false, a, false, b,
                                                (short)0, acc, false, false);
  }

  float* out = Y + (long long)(m0 + ((lane & 16) ? 8 : 0)) * CH + n0 + (lane & 15);
#pragma unroll
  for (int j = 0; j < 8; ++j) out[j * CH] = acc[j];
}

// ---------------------------------------------------------------------------
// Kernel 2: fill segmax with -inf (exact-size grid, b128 stores)
// ---------------------------------------------------------------------------
__global__ __launch_bounds__(256) void init_neginf_kernel(float4* __restrict__ S4, int n4) {
  int i = blockIdx.x * 256 + threadIdx.x;
  if (i < n4) {
    const float ni = -__builtin_inff();
    S4[i] = make_float4(ni, ni, ni, ni);
  }
}

// ---------------------------------------------------------------------------
// Kernel 3: scatter-max. One wave per edge, 4 channels per lane.
// float4 gather of y[row] (L2 hit), then 4 non-returning
// global_atomic_max_num_f32 at device scope (fire-and-forget, STOREcnt).
// ---------------------------------------------------------------------------
__global__ __launch_bounds__(256) void scatter_max_kernel(
    const long long* __restrict__ rows, const long long* __restrict__ cols,
    const float* __restrict__ Y, float* S, int E) {
  const long long gtid = (long long)blockIdx.x * 256 + threadIdx.x;
  const int edge = (int)(gtid >> 5);
  if (edge >= E) return;
  const int c0 = (threadIdx.x & 31) << 2;

  const int r = (int)rows[edge];
  const int c = (int)cols[edge];
  const float4 v = *(const float4*)(Y + (long long)r * CH + c0);
  float* p = S + (long long)c * CH + c0;

  asm volatile("global_atomic_max_num_f32 %0, %1, off scope:SCOPE_DEV"
               :: "v"(p + 0), "v"(v.x) : "memory");
  asm volatile("global_atomic_max_num_f32 %0, %1, off scope:SCOPE_DEV"
               :: "v"(p + 1), "v"(v.y) : "memory");
  asm volatile("global_atomic_max_num_f32 %0, %1, off scope:SCOPE_DEV"
               :: "v"(p + 2), "v"(v.z) : "memory");
  asm volatile("global_atomic_max_num_f32 %0, %1, off scope:SCOPE_DEV"
               :: "v"(p + 3), "v"(v.w) : "memory");
}

// ---------------------------------------------------------------------------
// Kernel 4: Out[m,p] = relu( sum_o M[m,o] * Wp[p,o] ),
//           M[m,o] = (row m empty) ? 0 : S[m,o] - Y[m,o]
// Y aliases Out (same buffer): every block reads only its own 16 rows in the
// k-loop, then __syncthreads() gates the stores -> no RAW hazard.
// ---------------------------------------------------------------------------
__global__ __launch_bounds__(256) void gemm_phi_kernel(
    const float* Yin, const float* __restrict__ S,
    const float* __restrict__ Wp, float* Out) {
  const int lane = threadIdx.x & 31;
  const int wave = threadIdx.x >> 5;
  const int m0 = blockIdx.x << 4;
  const int n0 = wave << 4;
  const int kb = (lane & 16) ? 2 : 0;

  const long long mrow = (long long)(m0 + (lane & 15)) * CH;
  const float* srow = S + mrow + kb;
  const float* yrow = Yin + mrow + kb;
  const float* brow = Wp + (long long)(n0 + (lane & 15)) * CH + kb;

  // Empty segment <=> channel 0 of this row is still -inf.
  const bool empty = (S[mrow] == -__builtin_inff());

  v8f acc = {};
#pragma unroll
  for (int kk = 0; kk < CH; kk += 4) {
    v2f s = *(const v2f*)(srow + kk);
    v2f y = *(const v2f*)(yrow + kk);
    v2f a;
    a.x = empty ? 0.0f : (s.x - y.x);
    a.y = empty ? 0.0f : (s.y - y.y);
    v2f b = *(const v2f*)(brow + kk);
    acc = __builtin_amdgcn_wmma_f32_16x16x4_f32(false, a, false, b,
                                                (short)0, acc, false, false);
  }

  __syncthreads();  // all waves done reading Yin rows before stores (alias)

  float* out = Out + (long long)(m0 + ((lane & 16) ? 8 : 0)) * CH + n0 + (lane & 15);
#pragma unroll
  for (int j = 0; j < 8; ++j) out[j * CH] = fmaxf(acc[j], 0.0f);
}

// ---------------------------------------------------------------------------
extern "C" void kernel_launch(void* const* d_in, const int* in_sizes, int n_in,
                              void* d_out, int out_size, void* d_ws, size_t ws_size,
                              hipStream_t stream) {
  const float* x  = (const float*)d_in[0];           // [N,128]
  const float* Wt = (const float*)d_in[1];           // [128,128]
  const float* Wp = (const float*)d_in[2];           // [128,128]
  const long long* edges = (const long long*)d_in[3];// [2,E] int64
  const int E = in_sizes[3] / 2;

  float* Y = (float*)d_out;  // y staged in d_out (51.2MB), overwritten at end
  float* S = (float*)d_ws;   // segmax, needs N*CH*4 = 51.2MB of workspace

  const int rowBlocks = NN / 16;          // 6250
  const int n4 = NN * CH / 4;             // 3.2M float4

  gemm_theta_kernel<<<rowBlocks, 256, 0, stream>>>(x, Wt, Y);
  init_neginf_kernel<<<(n4 + 255) / 256, 256, 0, stream>>>((float4*)S, n4);
  scatter_max_kernel<<<(int)(((long long)E * 32 + 255) / 256), 256, 0, stream>>>(
      edges, edges + E, Y, S, E);
  gemm_phi_kernel<<<rowBlocks, 256, 0, stream>>>(Y, S, Wp, (float*)d_out);
}